// OneDimEquivalent_51926154609381
// MI455X (gfx1250) — compile-verified
//
#include <hip/hip_runtime.h>
#include <cstdint>

// ---------------------------------------------------------------------------
// OneDimEquivalent scan for MI455X (gfx1250)
//
// Strategy:
//   * The per-step 128-node quadrature is a pure 1-D function g(delta):
//     tabulate it once (8192-entry LUT, delta in [0,16], 1/delta tail).
//   * Main kernel: one thread per batch row, sequential scan over T,
//     LUT in LDS, u tiles / z tiles moved with the Tensor Data Mover
//     (tensor_load_to_lds / tensor_store_from_lds), double buffered with
//     s_wait_tensorcnt. TDM LDS padding (1 dword per 32) kills bank conflicts.
// ---------------------------------------------------------------------------

#define NQ      128          // Gauss-Legendre nodes (matches reference)
#define A_INT   5.0          // quadrature interval [-A, A]
#define LUT_N   8192
#define DMAX    16.0f
#define T_LEN   4096
#define B_LEN   4096
#define TC      32           // time chunk per TDM tile
#define ROWS    256          // rows per block (= blockDim.x)
#define NC      (T_LEN / TC) // 128 chunks
#define UROW    33           // padded LDS row stride in dwords (TC + 1 pad)

typedef unsigned int u32x4 __attribute__((ext_vector_type(4)));
typedef int          i32x4 __attribute__((ext_vector_type(4)));
typedef int          i32x8 __attribute__((ext_vector_type(8)));

// ------------------- kernel 1: Gauss-Legendre nodes (Newton, f64) ----------
__global__ void gl_nodes_kernel(double* ws) {
  int i = threadIdx.x;                    // 0..63 -> positive roots
  if (i >= NQ / 2) return;
  const double pi = 3.14159265358979323846;
  double z = cos(pi * (i + 0.75) / (NQ + 0.5));
  double pp = 1.0;
  for (int it = 0; it < 64; ++it) {
    double p1 = 1.0, p2 = 0.0;
    for (int j = 1; j <= NQ; ++j) {
      double p3 = p2; p2 = p1;
      p1 = ((2.0 * j - 1.0) * z * p2 - (j - 1.0) * p3) / (double)j;
    }
    pp = NQ * (z * p1 - p2) / (z * z - 1.0);
    double z1 = z;
    z = z1 - p1 / pp;
    if (fabs(z - z1) < 1e-15) break;
  }
  double w  = 2.0 / ((1.0 - z * z) * pp * pp);
  double zn = A_INT * z;                       // node in [-A, A]
  double wf = A_INT * w * exp(-0.5 * zn * zn); // fold exp(-z^2/2) into weight
  ws[i]      = zn;
  ws[64 + i] = wf;
}

// ------------------- kernel 2: tabulate g(delta) in f64 --------------------
__global__ void lut_kernel(const double* __restrict__ ws,
                           float* __restrict__ lut) {
  int j = blockIdx.x * blockDim.x + threadIdx.x;
  if (j >= LUT_N) return;
  double delta = (double)j * ((double)DMAX / (double)(LUT_N - 1));
  double acc = 0.0;
  for (int i = 0; i < NQ / 2; ++i) {           // symmetric pairs: x2
    double t = tanh(delta * ws[i]);
    acc += ws[64 + i] * (1.0 - t * t);
  }
  const double inv_sqrt_2pi = 0.39894228040143267794;
  lut[j] = (float)(2.0 * acc * inv_sqrt_2pi);
}

// ------------------- TDM descriptor helpers --------------------------------
__device__ __forceinline__ u32x4 tdm_g0(uint32_t lds_off, const void* gptr) {
  uint64_t ga = (uint64_t)(uintptr_t)gptr;
  u32x4 g;
  g[0] = 1u;                                          // count=1, user mode
  g[1] = lds_off;                                     // LDS byte address
  g[2] = (uint32_t)ga;                                // global_addr[31:0]
  g[3] = ((uint32_t)(ga >> 32) & 0x01FFFFFFu)         // global_addr[56:32]
         | 0x80000000u;                               // type=2 ("image")
  return g;
}

__device__ __forceinline__ i32x8 tdm_zero8(void) {
  i32x8 z;
  z[0] = 0; z[1] = 0; z[2] = 0; z[3] = 0;
  z[4] = 0; z[5] = 0; z[6] = 0; z[7] = 0;
  return z;
}

// ------------------- kernel 3: the scan ------------------------------------
__global__ void scan_kernel(const float* __restrict__ u,
                            const float* __restrict__ lut_g,
                            float* __restrict__ out) {
  extern __shared__ char smem[];
  float* lut_s = (float*)(smem);                  //  32768 B : g LUT
  float* ub0   = (float*)(smem + 32768);          //  33792 B : u tile (padded)
  float* ub1   = (float*)(smem + 66560);          //  33792 B
  float* zb0   = (float*)(smem + 100352);         //  32768 B : z tile [TC][ROWS]
  float* zb1   = (float*)(smem + 133120);         //  32768 B   (total 165888)
  float* ubuf[2] = {ub0, ub1};
  float* zbuf[2] = {zb0, zb1};

  const int r  = threadIdx.x;
  const int b0 = blockIdx.x * ROWS;
  const int b  = b0 + r;
  const bool issuer = (threadIdx.x < 32);         // wave 0 drives the TDM

  // common group2/group3 (tensor_dim2.., zero strides)
  i32x4 g2l; g2l[0] = 1;  g2l[1] = 0; g2l[2] = 0; g2l[3] = 0;   // loads (<=2D)
  i32x4 g2s; g2s[0] = 64; g2s[1] = 0; g2s[2] = 0; g2s[3] = 0;   // store (3D, dim2>=TC)
  i32x4 g3;  g3[0] = 0;   g3[1] = 0;  g3[2] = 0;  g3[3] = 0;

  const i32x8 gx = tdm_zero8();                   // extra group (6-arg builtin form)

  // group1 for u tiles: 2D, 4B elems, tile 32x256, tensor stride 4096,
  // LDS padding: 1 dword after every 32 dwords -> row stride 33 dwords.
  i32x8 g1u;
  g1u[0] = (2 << 16) | (1 << 20) | (4 << 22);     // data_size=4B, pad_en, ivl=32dw, amt=1dw
  g1u[1] = (int)((unsigned)T_LEN << 16);          // tensor_dim0 = 4096
  g1u[2] = (int)((unsigned)B_LEN << 16);          // tensor_dim1 = 4096
  g1u[3] = (int)((unsigned)TC << 16);             // tile_dim0 = 32
  g1u[4] = ROWS;                                  // tile_dim1 = 256, tile_dim2 = 0
  g1u[5] = T_LEN;                                 // tensor_dim0_stride = 4096
  g1u[6] = 0; g1u[7] = 0;

  // ---- prologue: DMA LUT + first u tile into LDS --------------------------
  if (issuer) {
    i32x8 g1l;                                    // 1-D LUT tile, 8192 x f32
    g1l[0] = (2 << 16);
    g1l[1] = (int)((unsigned)LUT_N << 16);        // tensor_dim0 = 8192
    g1l[2] = (1 << 16);                           // tensor_dim1 = 1
    g1l[3] = (int)((unsigned)LUT_N << 16);        // tile_dim0 = 8192
    g1l[4] = 0; g1l[5] = LUT_N; g1l[6] = 0; g1l[7] = 0;
    __builtin_amdgcn_tensor_load_to_lds(
        tdm_g0((uint32_t)(uintptr_t)lut_s, lut_g), g1l, g2l, g3, gx, 0);
    __builtin_amdgcn_tensor_load_to_lds(
        tdm_g0((uint32_t)(uintptr_t)ub0, u + (size_t)b0 * T_LEN), g1u, g2l, g3, gx, 0);
    __builtin_amdgcn_s_wait_tensorcnt(0);
  }
  out[(size_t)b * (T_LEN + 1)] = 0.0f;            // z_hist[:, 0] = 0
  __syncthreads();

  // ---- main scan, double buffered ------------------------------------------
  float k = 0.0f, v = 0.0f;
  const float kScale = (float)(LUT_N - 1) / DMAX;

  for (int c = 0; c < NC; ++c) {
    const int cur = c & 1, nxt = cur ^ 1;

    if (issuer && (c + 1 < NC)) {                 // prefetch next u tile
      __builtin_amdgcn_tensor_load_to_lds(
          tdm_g0((uint32_t)(uintptr_t)ubuf[nxt],
                 u + (size_t)b0 * T_LEN + (size_t)(c + 1) * TC),
          g1u, g2l, g3, gx, 0);
    }

    // stage this thread's 32 u values into registers (b128 LDS loads, padded)
    float* ub = ubuf[cur];
    float* zb = zbuf[cur];
    float uu[TC];
#pragma unroll
    for (int j = 0; j < TC; ++j) uu[j] = ub[r * UROW + j];

#pragma unroll
    for (int j = 0; j < TC; ++j) {
      const float uj = uu[j];
      v += (uj - v) * 0.2f;                       // v += (-v + u) * dt/tau
      const float delta = sqrtf(k * k + uj * uj); // SIG_M = SIG_I = 1
      const float f = delta * kScale;
      float g;
      if (f < (float)(LUT_N - 1)) {               // lerp the LUT
        const int   i0 = (int)f;
        const float fr = f - (float)i0;
        const float ga = lut_s[i0], gb = lut_s[i0 + 1];
        g = ga + fr * (gb - ga);
      } else {                                    // asymptotic tail: g ~ 1/delta
        g = lut_s[LUT_N - 1] * ((float)(LUT_N - 1) / f);
      }
      k += (1.4f * g * k + 2.6f * g * v - k) * 0.2f;
      zb[j * ROWS + r] = 2.1f * k;                // column-major: conflict-free
    }
    __syncthreads();                              // zbuf[cur] complete

    if (issuer) {
      // 3-D store tile: X=1 elem, Y=256 rows @ stride 4097, Z=32 cols @ stride 1
      // LDS walk order (Z-major, Y-minor) matches zb[j*ROWS + r].
      i32x8 g1z;
      g1z[0] = (2 << 16);                          // 4B elems
      g1z[1] = (int)((unsigned)(T_LEN + 1) << 16); // tensor_dim0 = 4097
      g1z[2] = (int)((unsigned)B_LEN << 16);       // tensor_dim1 = 4096
      g1z[3] = (1 << 16);                          // tile_dim0 = 1
      g1z[4] = ROWS | (TC << 16);                  // tile_dim1=256, tile_dim2=32
      g1z[5] = T_LEN + 1;                          // tensor_dim0_stride = 4097
      g1z[6] = (1 << 16);                          // tensor_dim1_stride = 1
      g1z[7] = 0;
      __builtin_amdgcn_tensor_store_from_lds(
          tdm_g0((uint32_t)(uintptr_t)zb,
                 out + (size_t)b0 * (T_LEN + 1) + (size_t)c * TC + 1),
          g1z, g2s, g3, gx, 0);
      // in-order TENSORcnt: <=1 leaves only this store outstanding, so the
      // prefetched load (issued earlier) is guaranteed complete.
      if (c + 1 < NC) __builtin_amdgcn_s_wait_tensorcnt(1);
      else            __builtin_amdgcn_s_wait_tensorcnt(0);
    }
    __syncthreads();                              // ubuf[nxt] ready for all waves
  }
}

// ---------------------------------------------------------------------------
extern "C" void kernel_launch(void* const* d_in, const int* in_sizes, int n_in,
                              void* d_out, int out_size, void* d_ws, size_t ws_size,
                              hipStream_t stream) {
  (void)in_sizes; (void)n_in; (void)out_size; (void)ws_size;
  const float* u   = (const float*)d_in[0];
  float*       out = (float*)d_out;
  double*      wsd = (double*)d_ws;                      // 128 doubles: nodes+weights
  float*       lut = (float*)((char*)d_ws + 1024);       // 8192 floats

  gl_nodes_kernel<<<1, 64, 0, stream>>>(wsd);
  lut_kernel<<<LUT_N / 256, 256, 0, stream>>>(wsd, lut);

  const size_t shmem = 165888;                           // 162 KB < 320 KB/WGP
  scan_kernel<<<B_LEN / ROWS, ROWS, shmem, stream>>>(u, lut, out);
}